// RecognitionNetwork_88192858456174
// MI455X (gfx1250) — compile-verified
//
#include <hip/hip_runtime.h>
#include <stdint.h>

#define AS1 __attribute__((address_space(1)))
#define AS3 __attribute__((address_space(3)))
typedef int v4i32 __attribute__((vector_size(16)));  // matches builtin's pointee type

#if defined(__HIP_DEVICE_COMPILE__) && defined(__has_builtin)
#if __has_builtin(__builtin_amdgcn_global_load_async_to_lds_b128) && \
    __has_builtin(__builtin_amdgcn_s_wait_asynccnt)
#define USE_ASYNC_LDS 1
#endif
#endif

namespace {
constexpr int B = 16, T = 1024, V = 2048, L = 128;
constexpr int TP1 = T + 1, LP1 = L + 1;

// workspace layout (all offsets 256B aligned)
constexpr size_t OFF_ETOK = 0;                                   // B*T*L floats (8 MiB)
constexpr size_t OFF_EB   = OFF_ETOK + (size_t)B * T * L * 4;    // B*T floats (64 KiB)
constexpr size_t OFF_MASK = OFF_EB + (size_t)B * T * 4;          // B*T*L bytes (2 MiB)
constexpr size_t OFF_TST  = OFF_MASK + (size_t)B * T * L;        // B ints
} // namespace

// ---------------------------------------------------------------------------
// Backtrack (placed first so the disasm snippet shows the async-LDS path).
// One block per batch. Mask staged into LDS in 16KB chunks via async
// global->LDS DMA so the serial pointer-chase walk only pays LDS latency.
// ---------------------------------------------------------------------------
#define BT_THREADS 256
#define CHUNK 128  // time rows per LDS chunk

__global__ __launch_bounds__(BT_THREADS) void ctc_backtrack_kernel(
    const uint8_t* __restrict__ mask_bytes, const int* __restrict__ tokens,
    const int* __restrict__ tstart, float* __restrict__ alignment) {
  __shared__ uint4 lmask4[CHUNK * L / 16];  // 16 KiB
  __shared__ int ltok[L];
  uint8_t* lm = (uint8_t*)lmask4;

  const int b = blockIdx.x;
  const int tid = threadIdx.x;
  if (tid < L) ltok[tid] = tokens[b * L + tid];

  const uint8_t* mbase = mask_bytes + (size_t)b * T * L;
  const int ts = tstart[b];
  float* al = alignment + (size_t)b * T;
  int j = L;

  for (int ci = (T / CHUNK) - 1; ci >= 0; --ci) {
    const uint4* src = (const uint4*)(mbase + (size_t)ci * CHUNK * L);
#ifdef USE_ASYNC_LDS
    for (int i = tid; i < (CHUNK * L / 16); i += BT_THREADS) {
      __builtin_amdgcn_global_load_async_to_lds_b128(
          (AS1 v4i32*)(uintptr_t)(src + i),
          (AS3 v4i32*)(uint32_t)(uintptr_t)(lmask4 + i),
          0, 0);
    }
    __builtin_amdgcn_s_wait_asynccnt(0);
#else
    for (int i = tid; i < (CHUNK * L / 16); i += BT_THREADS) lmask4[i] = src[i];
#endif
    __syncthreads();
    if (tid == 0) {
      for (int t = (ci + 1) * CHUNK; t > ci * CHUNK; --t) {
        bool active = (t <= ts) && (j > 0);
        int tok = active ? ltok[j - 1] : 0;
        if (active && lm[(size_t)(t - 1 - ci * CHUNK) * L + (j - 1)]) --j;
        al[t - 1] = (float)tok;
      }
    }
    __syncthreads();
  }
}

// ---------------------------------------------------------------------------
// Gather: e_tok[b,t,j] = emission[b,t,tokens[b,j]], e_blank[b,t] = emission[b,t,0].
// 8 time steps per block so the token index is loaded once and 8 independent
// gathers stay in flight per thread.
// ---------------------------------------------------------------------------
#define TTILE 8

__global__ __launch_bounds__(128) void ctc_gather_kernel(
    const float* __restrict__ emission, const int* __restrict__ tokens,
    float* __restrict__ e_tok, float* __restrict__ e_blank) {
  const int blk = blockIdx.x;               // 0 .. B*(T/TTILE)-1
  const int b = blk / (T / TTILE);
  const int tbase = (blk % (T / TTILE)) * TTILE;
  const int j = threadIdx.x;                // 0 .. 127
  const int tok = tokens[b * L + j];
  const size_t bt0 = (size_t)b * T + tbase;
  const float* em = emission + bt0 * V;
#pragma unroll
  for (int k = 0; k < TTILE; ++k)
    e_tok[(bt0 + k) * L + j] = em[(size_t)k * V + tok];
  if (j < TTILE) e_blank[bt0 + j] = emission[(bt0 + j) * V];  // coalesced blanks
}

// ---------------------------------------------------------------------------
// Forward Viterbi scan. One wave32 per batch. Row of 129 values lives in
// registers (4 per lane + col0 on lane 0); shift-by-one column uses
// ds_bpermute (no barriers, no LDS in the serial chain).
// ---------------------------------------------------------------------------
__global__ __launch_bounds__(32) void ctc_forward_kernel(
    const float* __restrict__ e_tok, const float* __restrict__ e_blank,
    float* __restrict__ trellis,        // [B][1025][129]
    uint32_t* __restrict__ mask_words,  // [B][T][L] bytes viewed as u32
    int* __restrict__ tstart) {
  const int b = blockIdx.x;
  const int lane = threadIdx.x;         // 0..31 (wave32)
  const float NEG_INF = -__builtin_inff();
  const float POS_INF =  __builtin_inff();

  const float* et_base = e_tok + (size_t)b * T * L;
  const float* eb_base = e_blank + (size_t)b * T;
  float* tre = trellis + (size_t)b * TP1 * LP1;
  uint32_t* mw = mask_words + (size_t)b * T * (L / 4);

  const int cbase = 4 * lane + 1;       // columns cbase..cbase+3

  // row 0: [0, -inf x 128]
  float r0 = NEG_INF, r1 = NEG_INF, r2 = NEG_INF, r3 = NEG_INF;
  tre[cbase + 0] = r0; tre[cbase + 1] = r1;
  tre[cbase + 2] = r2; tre[cbase + 3] = r3;
  if (lane == 0) tre[0] = 0.0f;

  float col0 = 0.0f;                    // trellis[t][0] (current row)
  float csum = 0.0f;                    // running cumsum of blank emissions
  float best = NEG_INF;                 // running max of trellis[:,128] (row0=-inf)
  int bestT = 0;

  // depth-2 software pipeline on the coalesced e_tok row + scalar blank
  float4 eC = *(const float4*)(et_base + 0 * (size_t)L + 4 * lane);
  float ebC = eb_base[0];
  float4 eN = *(const float4*)(et_base + 1 * (size_t)L + 4 * lane);
  float ebN = eb_base[1];

  for (int t = 0; t < T; ++t) {
    float4 eNN; float ebNN;
    int t2 = t + 2;
    if (t2 < T) {
      eNN  = *(const float4*)(et_base + (size_t)t2 * L + 4 * lane);
      ebNN = eb_base[t2];
    } else { eNN = eN; ebNN = ebN; }
    if (t + 8 < T)  // warm WGP$/L0 well ahead of the pipeline (global_prefetch_b8)
      __builtin_prefetch(et_base + (size_t)(t + 8) * L + 4 * lane, 0, 3);

    // previous row's column (cbase-1): neighbor lane's r3 (or col0 for lane 0)
    float left = __int_as_float(
        __builtin_amdgcn_ds_bpermute((lane - 1) << 2, __float_as_int(r3)));
    if (lane == 0) left = col0;

    const float eb = ebC;
    float s0 = r0 + eb, s1 = r1 + eb, s2 = r2 + eb, s3 = r3 + eb;      // stayed
    float c0 = left + eC.x, c1 = r0 + eC.y, c2 = r1 + eC.z, c3 = r2 + eC.w; // changed
    uint32_t m = (uint32_t)(c0 > s0)        | ((uint32_t)(c1 > s1) << 8) |
                 ((uint32_t)(c2 > s2) << 16) | ((uint32_t)(c3 > s3) << 24);
    r0 = fmaxf(s0, c0); r1 = fmaxf(s1, c1);
    r2 = fmaxf(s2, c2); r3 = fmaxf(s3, c3);

    // column 0 of row t+1: cumsum, overwritten by +inf for the last L rows
    csum += eb;
    float v0 = (t + 1 >= TP1 - L) ? POS_INF : csum;

    float* trow = tre + (size_t)(t + 1) * LP1;
    if (lane == 0) trow[0] = v0;
    trow[cbase + 0] = r0; trow[cbase + 1] = r1;
    trow[cbase + 2] = r2; trow[cbase + 3] = r3;
    mw[(size_t)t * (L / 4) + lane] = m;

    col0 = v0;
    if (lane == 31 && r3 > best) { best = r3; bestT = t + 1; }  // first max kept

    eC = eN; ebC = ebN; eN = eNN; ebN = ebNN;
  }
  if (lane == 31) tstart[b] = bestT;
}

// ---------------------------------------------------------------------------
extern "C" void kernel_launch(void* const* d_in, const int* in_sizes, int n_in,
                              void* d_out, int out_size, void* d_ws, size_t ws_size,
                              hipStream_t stream) {
  (void)in_sizes; (void)n_in; (void)out_size; (void)ws_size;
  const float* emission = (const float*)d_in[0];  // [B,T,V] f32 log-softmax
  const int*   tokens   = (const int*)d_in[1];    // [B,L] i32

  float* alignment = (float*)d_out;               // [B,T] f32
  float* trellis   = (float*)d_out + (size_t)B * T;  // [B,1025,129] f32

  uint8_t* ws = (uint8_t*)d_ws;
  float*    e_tok   = (float*)(ws + OFF_ETOK);
  float*    e_blank = (float*)(ws + OFF_EB);
  uint32_t* mask_w  = (uint32_t*)(ws + OFF_MASK);
  const uint8_t* mask_b = (const uint8_t*)(ws + OFF_MASK);
  int*      tstart  = (int*)(ws + OFF_TST);

  ctc_gather_kernel<<<B * (T / TTILE), 128, 0, stream>>>(emission, tokens, e_tok, e_blank);
  ctc_forward_kernel<<<B, 32, 0, stream>>>(e_tok, e_blank, trellis, mask_w, tstart);
  ctc_backtrack_kernel<<<B, BT_THREADS, 0, stream>>>(mask_b, tokens, tstart, alignment);
}